// SpecAugment_80865644249532
// MI455X (gfx1250) — compile-verified
//
#include <hip/hip_runtime.h>
#include <stdint.h>

typedef __attribute__((ext_vector_type(4))) float v4f;

#define CLIPV 10.0f

// Per-(b,s) augmentation parameters, precomputed by the setup kernel into d_ws.
struct BSParams {
  int t0s, t0e, t1s, t1e;   // two time-mask intervals [start,end) ; -1,-1 = not applied
  int f0s, f0e, f1s, f1e;   // two freq-mask intervals
  float scale;              // random scale (1.0 if not applied)
  float amp;                // noise amplitude (0.06 or 0.0)
  float on;                 // sensor_on (1.0 / 0.0) from the input mask
  float pad;
};

static constexpr int THREADS = 256;
static constexpr int ITERS   = 8;          // float4 chunks per block per thread
static constexpr int B_ = 32, S_ = 8, T_ = 1024, F_ = 128;
static constexpr int TOTAL_F4 = B_ * S_ * T_ * F_ / 4;       // 8,388,608
static constexpr int F4_PER_BS = T_ * F_ / 4;                // 32,768 (power of two)

__device__ __forceinline__ uint32_t hash32(uint32_t x) {
  x ^= x >> 17; x *= 0xed5ad4bbu;
  x ^= x >> 11; x *= 0xac4c1b51u;
  x ^= x >> 15; x *= 0x31848babu;
  x ^= x >> 14;
  return x;
}
__device__ __forceinline__ float u01(uint32_t h) {
  return (float)(h >> 8) * (1.0f / 16777216.0f);   // [0,1)
}

// Async global->LDS copy of one float4 per lane (CDNA5 ASYNCcnt path).
// VDST operand: per the ISA aperture layout, the low 32 bits of a flat LDS
// address ARE the LDS byte offset, so trunc(ptrtoint(lds_slot)) is the correct
// operand value. The ptrtoint also CAPTURES `stage`, which together with the
// "memory" clobber keeps the later ds_load_b128 read-backs alive and ordered.
// (No "m" operands: the AMDGPU backend cannot select LDS asm memory operands.)
__device__ __forceinline__ void async_copy_f4(v4f* lds_slot, const v4f* gsrc) {
  asm volatile("global_load_async_to_lds_b128 %0, %1, off"
               :
               : "v"((uint32_t)(uintptr_t)lds_slot), "v"(gsrc)
               : "memory");
}

// ---------------- setup: 256 threads, one per (b,s) ----------------
__global__ __launch_bounds__(256) void specaug_setup(const int* __restrict__ mask,
                                                     BSParams* __restrict__ P) {
  const uint32_t tid = threadIdx.x;          // tid = b*8 + s
  if (tid >= 256) return;

  auto rng = [&](uint32_t stream) -> uint32_t {
    return hash32(tid * 0x9E3779B9u ^ (stream + 1u) * 0x85EBCA6Bu ^ 0x5EEDC0DEu);
  };
  auto axis_interval = [&](int dim, int maxw, uint32_t sbase, int& st, int& en) {
    int width = 1 + (int)(rng(sbase) % (uint32_t)maxw);      // [1, maxw]
    float u   = u01(rng(sbase + 1));
    int span  = dim - width; if (span < 1) span = 1;
    int start = (int)(u * (float)span);
    bool apply = u01(rng(sbase + 2)) > 0.35f;                // applied w.p. 0.65
    if (apply) { st = start; en = start + width; }
    else       { st = -1;    en = -1; }
  };

  BSParams p;
  axis_interval(T_, 122, 0,  p.t0s, p.t0e);   // max_t = int(1024*0.12) = 122
  axis_interval(T_, 122, 4,  p.t1s, p.t1e);
  axis_interval(F_, 15,  8,  p.f0s, p.f0e);   // max_f = int(128*0.12) = 15
  axis_interval(F_, 15,  12, p.f1s, p.f1e);

  p.amp = (u01(rng(100)) > 0.45f) ? 0.06f : 0.0f;            // noise w.p. 0.55
  float sc = 0.85f + 0.30f * u01(rng(101));
  p.scale  = (u01(rng(102)) > 0.5f) ? sc : 1.0f;             // scale w.p. 0.5
  p.on  = (mask[tid] > 0) ? 1.0f : 0.0f;
  p.pad = 0.0f;
  P[tid] = p;
}

// ---------------- main streaming kernel ----------------
// One block = 2048 contiguous float4 = 1/16 of one (b,s) slice -> params uniform.
// Double-buffered async global->LDS staging (ASYNCcnt), ds_load_b128 read-back,
// non-temporal b128 stores (256 MiB working set > 192 MB L2).
__global__ __launch_bounds__(256) void specaug_main(const float* __restrict__ specs,
                                                    const BSParams* __restrict__ P,
                                                    float* __restrict__ out) {
  __shared__ v4f stage[2 * THREADS];         // 8 KB, two 4 KB buffers

  const uint32_t tid        = threadIdx.x;
  const uint32_t blockStart = (uint32_t)blockIdx.x * (uint32_t)(ITERS * THREADS);
  const uint32_t bs         = blockStart / (uint32_t)F4_PER_BS;   // uniform per block

  const BSParams p = P[bs];                  // wave-uniform -> scalar loads
  const bool on      = p.on > 0.5f;
  const bool doNoise = p.amp != 0.0f;

  const v4f* gin  = (const v4f*)specs + blockStart + tid;
  v4f*       gout = (v4f*)out        + blockStart + tid;

  // Prologue: issue chunk 0 -> buf0, chunk 1 -> buf1 (ASYNCcnt = 2)
  async_copy_f4(&stage[tid],           gin);
  async_copy_f4(&stage[THREADS + tid], gin + THREADS);

#pragma unroll
  for (int i = 0; i < ITERS; ++i) {
    // Wait until chunk i has landed in LDS (one younger chunk may stay in flight).
    if (i + 1 < ITERS) asm volatile("s_wait_asynccnt 0x1" ::: "memory");
    else               asm volatile("s_wait_asynccnt 0x0" ::: "memory");

    const v4f x = stage[(uint32_t)(i & 1) * THREADS + tid];                // ds_load_b128

    const uint32_t g   = blockStart + (uint32_t)i * THREADS + tid;         // float4 index
    const uint32_t rem = g & (uint32_t)(F4_PER_BS - 1);
    const int      t   = (int)(rem >> 5);          // 32 float4 per T row
    const int      f0  = (int)((rem & 31u) << 2);  // first of 4 freq bins

    const float tm = (((t >= p.t0s) & (t < p.t0e)) | ((t >= p.t1s) & (t < p.t1e)))
                         ? 0.0f : 1.0f;

    float n[4] = {0.f, 0.f, 0.f, 0.f};
    if (doNoise) {   // uniform branch per block
      // Box-Muller, two pairs per float4. VALU-trimmed RNG: two full hash
      // chains for the radius uniforms, one LCG step each for the angles
      // (keeps compute safely below the HBM roofline floor).
      uint32_t hA = hash32(g * 4u + 0xB5297A4Du);
      uint32_t hC = hash32(g * 4u + 0x7FEB352Du);
      uint32_t hB = hA * 1664525u + 1013904223u;   // cheap derived angle bits
      uint32_t hD = hC * 1664525u + 362437u;
      float r0 = __fsqrt_rn(-2.0f * __logf(((hA >> 8) + 1u) * (1.0f / 16777216.0f)));
      float a0 = 6.2831853f * u01(hB);
      float r1 = __fsqrt_rn(-2.0f * __logf(((hC >> 8) + 1u) * (1.0f / 16777216.0f)));
      float a1 = 6.2831853f * u01(hD);
      n[0] = r0 * __cosf(a0);  n[1] = r0 * __sinf(a0);
      n[2] = r1 * __cosf(a1);  n[3] = r1 * __sinf(a1);
    }

    v4f r;
#pragma unroll
    for (int j = 0; j < 4; ++j) {
      const int fj = f0 + j;
      const float fm = (((fj >= p.f0s) & (fj < p.f0e)) | ((fj >= p.f1s) & (fj < p.f1e)))
                           ? 0.0f : 1.0f;
      float v = x[j] * tm * fm;
      v = (v + p.amp * n[j]) * p.scale;      // scale applies to noise too (matches ref)
      v = fminf(fmaxf(v, -CLIPV), CLIPV);
      r[j] = on ? v : x[j];
    }

    // Streaming store: non-temporal, working set exceeds L2.
    __builtin_nontemporal_store(r, gout + i * THREADS);

    // Refill the buffer we just consumed with chunk i+2. The ds_load above is a
    // memory read and cannot sink past this "memory"-clobbering asm.
    if (i + 2 < ITERS) {
      async_copy_f4(&stage[(uint32_t)(i & 1) * THREADS + tid],
                    gin + (i + 2) * THREADS);
    }
  }
}

extern "C" void kernel_launch(void* const* d_in, const int* in_sizes, int n_in,
                              void* d_out, int out_size, void* d_ws, size_t ws_size,
                              hipStream_t stream) {
  const float* specs = (const float*)d_in[0];   // [32,8,1024,128,1] f32
  const int*   mask  = (const int*)d_in[1];     // [32,8] i32
  BSParams*    P     = (BSParams*)d_ws;         // 256 * 48 B = 12 KB scratch

  specaug_setup<<<1, 256, 0, stream>>>(mask, P);

  const int blocks = TOTAL_F4 / (ITERS * THREADS);   // 4096
  specaug_main<<<blocks, THREADS, 0, stream>>>(specs, P, (float*)d_out);
}